// RNN_76682346103210
// MI455X (gfx1250) — compile-verified
//
#include <hip/hip_runtime.h>
#include <math.h>

#define DEV __device__ __forceinline__

typedef __attribute__((ext_vector_type(2))) float f32x2;
typedef __attribute__((ext_vector_type(8))) float f32x8;

constexpr int kN = 64;
constexpr int kT = 512;
constexpr int kD = 1024;
constexpr int kH = 1024;

constexpr int LDSA = 36;    // A stride (floats): 144B rows, conflict-free b64 frags
constexpr int LDSB = 1036;  // resident-B stride: 1036%64=12 -> 12n%64 distinct, n=0..15
constexpr int kScanBlocks = kH / 16;  // 64 persistent blocks

DEV f32x8 wmma4(f32x2 a, f32x2 b, f32x8 c) {
  return __builtin_amdgcn_wmma_f32_16x16x4_f32(false, a, false, b, (short)0, c,
                                               false, false);
}

// Copy 16 contiguous floats (one row chunk) global -> LDS.
DEV void stage_row16(float* dst, const float* src) {
  const float4* s = (const float4*)src;
  float4 v0 = s[0], v1 = s[1], v2 = s[2], v3 = s[3];
  float4* d = (float4*)dst;
  d[0] = v0; d[1] = v1; d[2] = v2; d[3] = v3;
}

// Gather CNT elements of one weight column (stride ldg) -> contiguous LDS.
template <int CNT>
DEV void stage_col(float* dst, const float* src, int ldg) {
  float tmp[CNT];
#pragma unroll
  for (int i = 0; i < CNT; ++i) tmp[i] = src[(size_t)i * ldg];
#pragma unroll
  for (int i = 0; i < CNT / 4; ++i) {
    float4 v; v.x = tmp[4*i]; v.y = tmp[4*i+1]; v.z = tmp[4*i+2]; v.w = tmp[4*i+3];
    ((float4*)dst)[i] = v;
  }
}

// ---------------------------------------------------------------------------
// Phase 1: out[m, :] = x[m, :] @ Wx + b  (M = N*T = 32768), exact f32 WMMA.
// Block: 256 threads (8 waves), tile 128(M) x 64(H), double-buffered K=32.
// ---------------------------------------------------------------------------
__global__ __launch_bounds__(256, 2) void rnn_xw_gemm(
    const float* __restrict__ x, const float* __restrict__ Wx,
    const float* __restrict__ bias, float* __restrict__ out) {
  __shared__ float As[2][128 * LDSA];  // [row][k]
  __shared__ float Bs[2][64 * LDSA];   // [col][k] (transposed)

  const int tid  = threadIdx.x;
  const int wave = tid >> 5;
  const int lane = tid & 31;
  const int g = lane >> 4, n = lane & 15;
  const int rblock = blockIdx.y * 128;
  const int cblock = blockIdx.x * 64;

  const int ar = tid >> 1, ak = (tid & 1) * 16;  // A: 128 rows, 16 k each
  const int bc = tid & 63, bk = (tid >> 6) * 8;  // B: 64 cols, 8 k each

  const float* aSrc = x + (size_t)(rblock + ar) * kD + ak;
  const float* bSrc = Wx + (size_t)bk * kH + cblock + bc;

  stage_row16(&As[0][ar * LDSA + ak], aSrc);
  stage_col<8>(&Bs[0][bc * LDSA + bk], bSrc, kH);
  __syncthreads();

  f32x8 acc[4] = {};
  int buf = 0;
  for (int kk = 0; kk < kD; kk += 32) {
    if (kk + 32 < kD) {  // prefetch next chunk into the other buffer
      stage_row16(&As[buf ^ 1][ar * LDSA + ak], aSrc + kk + 32);
      stage_col<8>(&Bs[buf ^ 1][bc * LDSA + bk], bSrc + (size_t)(kk + 32) * kH, kH);
    }
    const float* ap = &As[buf][(wave * 16 + n) * LDSA + 2 * g];
#pragma unroll
    for (int j = 0; j < 8; ++j) {
      f32x2 a = *(const f32x2*)(ap + 4 * j);
#pragma unroll
      for (int f = 0; f < 4; ++f) {
        f32x2 b = *(const f32x2*)(&Bs[buf][(f * 16 + n) * LDSA + 4 * j + 2 * g]);
        acc[f] = wmma4(a, b, acc[f]);
      }
    }
    __syncthreads();
    buf ^= 1;
  }

#pragma unroll
  for (int f = 0; f < 4; ++f) {
    const int col = cblock + f * 16 + n;
    const float bv = bias[col];
    float* orow = out + (size_t)(rblock + wave * 16 + 8 * g) * kH + col;
#pragma unroll
    for (int r = 0; r < 8; ++r)
      orow[(size_t)r * kH] = acc[f][r] + bv;  // D row = r + 8g
  }
}

// ---------------------------------------------------------------------------
// Phase 2: persistent scan kernel. 64 blocks (one 16-col slice of H each),
// 128 threads (4 waves x 16 rows). Wh slice (1024x16 = 64KB) stays resident
// in LDS for all 512 steps; h chunks double-buffer through LDS; steps are
// separated by a device-wide generation barrier in d_ws.
// ---------------------------------------------------------------------------
__global__ __launch_bounds__(128) void rnn_scan(
    const float* __restrict__ h0, const float* __restrict__ Wh,
    float* __restrict__ out, unsigned* __restrict__ bar_cnt,
    unsigned* __restrict__ bar_gen) {
  __shared__ float Bw[16 * LDSB];     // resident Wh slice: [col][k], 66.3 KB
  __shared__ float As[2][64 * LDSA];  // h_{t-1} K-chunk double buffer, 18.4 KB

  const int tid  = threadIdx.x;
  const int wave = tid >> 5;          // rows wave*16 .. wave*16+15
  const int lane = tid & 31;
  const int g = lane >> 4, n = lane & 15;
  const int cblock = blockIdx.x * 16;

  {  // one-time: load this block's Wh slice (transposed) into LDS
    const int c = tid & 15, q = tid >> 4;  // 8 threads per column, 128 k each
#pragma unroll 1
    for (int i = 0; i < 128; i += 8)
      stage_col<8>(&Bw[c * LDSB + q * 128 + i],
                   Wh + (size_t)(q * 128 + i) * kH + cblock + c, kH);
  }

  const int ar = tid >> 1, ak = (tid & 1) * 16;  // A staging: 64 rows, 16 k each
  const int col = cblock + n;

  for (int t = 0; t < kT; ++t) {
    const float* hsrc;
    long long hstr;
    if (t == 0) { hsrc = h0;                        hstr = kH; }
    else        { hsrc = out + (size_t)(t - 1) * kH; hstr = (long long)kT * kH; }
    const float* aSrc = hsrc + (size_t)ar * hstr + ak;

    stage_row16(&As[0][ar * LDSA + ak], aSrc);
    __syncthreads();  // also covers Bw prologue at t==0

    f32x8 acc[4] = {};
    int buf = 0;
    for (int kk = 0; kk < kH; kk += 32) {
      if (kk + 32 < kH)
        stage_row16(&As[buf ^ 1][ar * LDSA + ak], aSrc + kk + 32);
      const float* ap = &As[buf][(wave * 16 + n) * LDSA + 2 * g];
      const float* bp = &Bw[n * LDSB + kk + 2 * g];
#pragma unroll
      for (int j = 0; j < 8; ++j) {
        f32x2 a = *(const f32x2*)(ap + 4 * j);
        f32x2 b = *(const f32x2*)(bp + 4 * j);
        acc[j & 3] = wmma4(a, b, acc[j & 3]);  // 4-way rotation: chain depth /4
      }
      __syncthreads();
      buf ^= 1;
    }

    const f32x8 s = (acc[0] + acc[1]) + (acc[2] + acc[3]);
#pragma unroll
    for (int r = 0; r < 8; ++r) {
      const int row = wave * 16 + r + 8 * g;                 // batch index
      const size_t idx = ((size_t)row * kT + t) * kH + col;  // (N,T,H) flat
      out[idx] = tanhf(s[r] + out[idx]);  // same thread reads xw & writes h
    }

    // ---- device-wide step barrier (monotonic generation) ----
    __threadfence();   // make this thread's h stores device-visible
    __syncthreads();   // whole block arrived & fenced
    if (tid == 0) {
      const unsigned target = (unsigned)(t + 1);
      unsigned prev = __hip_atomic_fetch_add(bar_cnt, 1u, __ATOMIC_ACQ_REL,
                                             __HIP_MEMORY_SCOPE_AGENT);
      if (prev == target * (unsigned)kScanBlocks - 1u) {
        __hip_atomic_store(bar_gen, target, __ATOMIC_RELEASE,
                           __HIP_MEMORY_SCOPE_AGENT);
      } else {
        while (__hip_atomic_load(bar_gen, __ATOMIC_ACQUIRE,
                                 __HIP_MEMORY_SCOPE_AGENT) < target)
          __builtin_amdgcn_s_sleep(1);
      }
    }
    __syncthreads();
  }
}

__global__ void init_barrier(unsigned* p) { p[0] = 0u; p[1] = 0u; }

extern "C" void kernel_launch(void* const* d_in, const int* in_sizes, int n_in,
                              void* d_out, int out_size, void* d_ws,
                              size_t ws_size, hipStream_t stream) {
  (void)in_sizes; (void)n_in; (void)out_size; (void)ws_size;
  const float* x  = (const float*)d_in[0];   // (N, T, D)
  const float* h0 = (const float*)d_in[1];   // (N, H)
  const float* Wx = (const float*)d_in[2];   // (D, H)
  const float* Wh = (const float*)d_in[3];   // (H, H)
  const float* b  = (const float*)d_in[4];   // (H,)
  float* out = (float*)d_out;                // (N, T, H)
  unsigned* bar = (unsigned*)d_ws;           // [cnt, gen]

  init_barrier<<<1, 1, 0, stream>>>(bar);

  // Phase 1: fill d_out with xw = x@Wx + b.
  dim3 g1(kH / 64, (kN * kT) / 128);
  rnn_xw_gemm<<<g1, 256, 0, stream>>>(x, Wx, b, out);

  // Phase 2: one persistent kernel runs all 512 steps in-place on d_out.
  rnn_scan<<<kScanBlocks, 128, 0, stream>>>(h0, Wh, out, bar, bar + 1);
}